// HAN_NC_2430951489937
// MI455X (gfx1250) — compile-verified
//
#include <hip/hip_runtime.h>
#include <hip/hip_bf16.h>
#include <math.h>

// ---------------------------------------------------------------------------
// HAN forward for MI455X (gfx1250, wave32).
//   h_paper  = x_paper  @ W_paper  + b   -> [50000,8,16]
//   h_author = x_author @ W_author + b   -> [20000,8,16]
//   out_writes = rel_conv(h_author -> h_paper, writes edges)   [50000,128]
//   out_cites  = rel_conv(h_paper  -> h_paper, cites  edges)   [50000,128]
//   (written relation + author_out are dead code -> skipped)
//   semantic attention over {writes, cites}; final @ lin_W[128,64] + lin_b
// All dense GEMMs use V_WMMA_F32_16X16X4_F32 (full f32 precision).
// ---------------------------------------------------------------------------

#define N_PAPER  50000
#define N_AUTHOR 20000
#define HID      128
#define EMB      64
#define NEG_SLOPE 0.2f

typedef float v2f __attribute__((ext_vector_type(2)));
typedef float v8f __attribute__((ext_vector_type(8)));

// ---------------- WMMA f32 16x16xK tile (K multiple of 4) -----------------
// A: row-major [16 x K] slab at `A` with leading dim lda  (element (r, k))
// B: row-major [K x 16] slab at `B` with leading dim ldb  (element (k, c))
// Lane mapping (ISA 7.12.2, 32-bit A 16x4): lane<16 -> K=k,k+1 ; lane>=16 -> K=k+2,k+3
// D: acc[j] = element (row = j + 8*(lane>=16), col = lane&15)
__device__ __forceinline__ v8f wmma_tile_f32(const float* __restrict__ A, int lda,
                                             const float* __restrict__ B, int ldb,
                                             int K, int lane) {
  const int r    = lane & 15;
  const int half = lane >> 4;
  const float* arow = A + r * lda + 2 * half;        // (r, 2*half + k)
  const float* bcol = B + (2 * half) * ldb + r;      // (2*half + k, r)
  v8f acc = {0.f, 0.f, 0.f, 0.f, 0.f, 0.f, 0.f, 0.f};
  for (int k = 0; k < K; k += 4) {
    v2f a = *(const v2f*)(arow + k);                 // 8B aligned: r*lda, k even
    v2f b;
    b.x = bcol[k * ldb];
    b.y = bcol[(k + 1) * ldb];
    acc = __builtin_amdgcn_wmma_f32_16x16x4_f32(
        /*neg_a=*/false, a, /*neg_b=*/false, b,
        /*c_mod=*/(short)0, acc, /*reuse_a=*/false, /*reuse_b=*/false);
  }
  return acc;
}

// ------------------- projection GEMM: H = X @ W + bias --------------------
// X:[M,128] W:[128,128] H:[M,128].  grid = M/16 blocks, 256 thr (8 waves),
// wave w computes the (blockIdx, w) 16x16 tile.
__global__ void proj_gemm(const float* __restrict__ X, const float* __restrict__ W,
                          const float* __restrict__ bias, float* __restrict__ H) {
  const int wave = threadIdx.x >> 5, lane = threadIdx.x & 31;
  const int mt = blockIdx.x, nt = wave;
  v8f acc = wmma_tile_f32(X + mt * 16 * HID, HID, W + nt * 16, HID, HID, lane);
  const int r = lane & 15, half = lane >> 4;
  const float bv = bias[nt * 16 + r];
#pragma unroll
  for (int j = 0; j < 8; ++j)
    H[(mt * 16 + j + 8 * half) * HID + nt * 16 + r] = acc[j] + bv;
}

// -------- per-node attention logits: a[n,h] = sum_d H[n,h,d]*att[h,d] -----
__global__ void node_att(const float* __restrict__ H, const float* __restrict__ att,
                         float* __restrict__ a, int total /* = N*8 */) {
  int i = blockIdx.x * blockDim.x + threadIdx.x;
  if (i >= total) return;
  const int h = i & 7;
  const float* hp = H + (size_t)i * 16;   // n*128 + h*16 == i*16
  const float* ap = att + h * 16;
  float s = 0.f;
#pragma unroll
  for (int d = 0; d < 16; ++d) s += hp[d] * ap[d];
  a[i] = s;
}

// ----------------------- float atomic max (sign-split) --------------------
__device__ __forceinline__ void atomicMaxF(float* addr, float v) {
  if (v >= 0.f) atomicMax((int*)addr, __float_as_int(v));
  else          atomicMin((unsigned int*)addr, __float_as_uint(v));
}

__device__ __forceinline__ float lrelu(float x) { return x > 0.f ? x : NEG_SLOPE * x; }

// --------------------------- edge pass 1: max -----------------------------
__global__ void edge_max(const float* __restrict__ as, const float* __restrict__ ad,
                         const int* __restrict__ src, const int* __restrict__ dst,
                         float* __restrict__ maxb, int E) {
  int e = blockIdx.x * blockDim.x + threadIdx.x;
  if (e >= E) return;
  const int s = src[e], d = dst[e];
#pragma unroll
  for (int h = 0; h < 8; ++h)
    atomicMaxF(&maxb[d * 8 + h], lrelu(as[s * 8 + h] + ad[d * 8 + h]));
}

// ------------------------ edge pass 2: sum of exp -------------------------
__global__ void edge_sum(const float* __restrict__ as, const float* __restrict__ ad,
                         const int* __restrict__ src, const int* __restrict__ dst,
                         const float* __restrict__ maxb, float* __restrict__ denom, int E) {
  int e = blockIdx.x * blockDim.x + threadIdx.x;
  if (e >= E) return;
  const int s = src[e], d = dst[e];
#pragma unroll
  for (int h = 0; h < 8; ++h) {
    float m = maxb[d * 8 + h];
    if (!__builtin_isfinite(m)) m = 0.f;
    atomicAdd(&denom[d * 8 + h], __expf(lrelu(as[s * 8 + h] + ad[d * 8 + h]) - m));
  }
}

// ----------- edge pass 3: weighted scatter (one wave per edge) ------------
// lane handles components c = lane + 32*i (i=0..3); head = c>>4.
__global__ void edge_scatter(const float* __restrict__ as, const float* __restrict__ ad,
                             const int* __restrict__ src, const int* __restrict__ dst,
                             const float* __restrict__ maxb, const float* __restrict__ denom,
                             const float* __restrict__ hsrc, float* __restrict__ out, int E) {
  const int wid  = blockIdx.x * (blockDim.x >> 5) + (threadIdx.x >> 5);
  const int lane = threadIdx.x & 31;
  if (wid >= E) return;
  const int s = src[wid], d = dst[wid];
#pragma unroll
  for (int i = 0; i < 4; ++i) {
    const int c = lane + 32 * i;
    const int h = c >> 4;
    float m = maxb[d * 8 + h];
    if (!__builtin_isfinite(m)) m = 0.f;
    const float e = __expf(lrelu(as[s * 8 + h] + ad[d * 8 + h]) - m);
    const float w = e / (denom[d * 8 + h] + 1e-16f);
    atomicAdd(&out[(size_t)d * HID + c], hsrc[(size_t)s * HID + c] * w);
  }
}

__global__ void relu_inplace(float* __restrict__ x, int n) {
  int i = blockIdx.x * blockDim.x + threadIdx.x;
  if (i < n) x[i] = fmaxf(x[i], 0.f);
}

// ------- semantic GEMM: colsum += sum_m tanh((OUT @ kW + kb)[m, :]) -------
__global__ void sem_gemm(const float* __restrict__ OUT, const float* __restrict__ kW,
                         const float* __restrict__ kb, float* __restrict__ colsum) {
  const int wave = threadIdx.x >> 5, lane = threadIdx.x & 31;
  const int mt = blockIdx.x, nt = wave;
  v8f acc = wmma_tile_f32(OUT + mt * 16 * HID, HID, kW + nt * 16, HID, HID, lane);
  const int r = lane & 15, half = lane >> 4;
  const float kbv = kb[nt * 16 + r];
  float p = 0.f;
#pragma unroll
  for (int j = 0; j < 8; ++j) p += tanhf(acc[j] + kbv);
  (void)half;
  const float po = __shfl_down(p, 16, 32);   // lane & lane+16 share a column
  if ((lane >> 4) == 0) atomicAdd(&colsum[nt * 16 + r], p + po);
}

// ----- semantic softmax over the two relation scores (single wave) --------
__global__ void sem_score(const float* __restrict__ q, const float* __restrict__ csw,
                          const float* __restrict__ csc, float* __restrict__ attn) {
  const int lane = threadIdx.x;
  float sw = 0.f, sc = 0.f;
  for (int f = lane; f < HID; f += 32) { sw += q[f] * csw[f]; sc += q[f] * csc[f]; }
#pragma unroll
  for (int off = 16; off > 0; off >>= 1) {
    sw += __shfl_down(sw, off, 32);
    sc += __shfl_down(sc, off, 32);
  }
  if (lane == 0) {
    sw *= (1.f / (float)N_PAPER);
    sc *= (1.f / (float)N_PAPER);
    const float mx = fmaxf(sw, sc);
    const float ew = __expf(sw - mx), ec = __expf(sc - mx);
    const float inv = 1.f / (ew + ec);
    attn[0] = ew * inv;
    attn[1] = ec * inv;
  }
}

// -- final GEMM: d_out = (a0*out_w + a1*out_c) @ lin_W + lin_b  [50000,64] --
__global__ void final_gemm(const float* __restrict__ ow, const float* __restrict__ oc,
                           const float* __restrict__ attn, const float* __restrict__ linW,
                           const float* __restrict__ linb, float* __restrict__ out) {
  const int wave = threadIdx.x >> 5, lane = threadIdx.x & 31;
  const int tile = blockIdx.x * 8 + wave;              // wave-uniform
  if (tile >= (N_PAPER / 16) * (EMB / 16)) return;
  const int mt = tile >> 2, nt = tile & 3;
  const float a0 = attn[0], a1 = attn[1];
  const int r = lane & 15, half = lane >> 4;
  const float* aw = ow + (size_t)(mt * 16 + r) * HID + 2 * half;
  const float* ac = oc + (size_t)(mt * 16 + r) * HID + 2 * half;
  const float* bcol = linW + (2 * half) * EMB + nt * 16 + r;
  v8f acc = {0.f, 0.f, 0.f, 0.f, 0.f, 0.f, 0.f, 0.f};
  for (int k = 0; k < HID; k += 4) {
    v2f a, b;
    a.x = a0 * aw[k]     + a1 * ac[k];
    a.y = a0 * aw[k + 1] + a1 * ac[k + 1];
    b.x = bcol[k * EMB];
    b.y = bcol[(k + 1) * EMB];
    acc = __builtin_amdgcn_wmma_f32_16x16x4_f32(false, a, false, b, (short)0, acc,
                                                false, false);
  }
  const float bv = linb[nt * 16 + r];
#pragma unroll
  for (int j = 0; j < 8; ++j)
    out[(size_t)(mt * 16 + j + 8 * half) * EMB + nt * 16 + r] = acc[j] + bv;
}

// ---------------------------------------------------------------------------
extern "C" void kernel_launch(void* const* d_in, const int* in_sizes, int n_in,
                              void* d_out, int out_size, void* d_ws, size_t ws_size,
                              hipStream_t stream) {
  const float* x_paper   = (const float*)d_in[0];
  const float* x_author  = (const float*)d_in[1];
  const float* W_paper   = (const float*)d_in[2];
  const float* b_paper   = (const float*)d_in[3];
  const float* W_author  = (const float*)d_in[4];
  const float* b_author  = (const float*)d_in[5];
  const float* att_s_w   = (const float*)d_in[6];
  const float* att_d_w   = (const float*)d_in[7];
  // d_in[8], d_in[9]: written relation (dead code)
  const float* att_s_c   = (const float*)d_in[10];
  const float* att_d_c   = (const float*)d_in[11];
  const float* k_W       = (const float*)d_in[12];
  const float* k_b       = (const float*)d_in[13];
  const float* q         = (const float*)d_in[14];
  const float* lin_W     = (const float*)d_in[15];
  const float* lin_b     = (const float*)d_in[16];
  const int*   writes_src = (const int*)d_in[17];
  const int*   writes_dst = (const int*)d_in[18];
  // d_in[19], d_in[20]: written edges (dead code)
  const int*   cites_src  = (const int*)d_in[21];
  const int*   cites_dst  = (const int*)d_in[22];
  const int E_W = in_sizes[17];
  const int E_C = in_sizes[21];

  // ---------------- workspace carve-out (aligned to 256B) -----------------
  char* p = (char*)d_ws;
  auto carve = [&](size_t nfloats) {
    float* r = (float*)p;
    p += (nfloats * sizeof(float) + 255) & ~(size_t)255;
    return r;
  };
  float* h_paper  = carve((size_t)N_PAPER * HID);
  float* h_author = carve((size_t)N_AUTHOR * HID);
  float* out_w    = carve((size_t)N_PAPER * HID);
  float* out_c    = carve((size_t)N_PAPER * HID);
  float* a_s_wr   = carve((size_t)N_AUTHOR * 8);
  float* a_d_wr   = carve((size_t)N_PAPER * 8);
  float* a_s_ci   = carve((size_t)N_PAPER * 8);
  float* a_d_ci   = carve((size_t)N_PAPER * 8);
  float* maxb     = carve((size_t)N_PAPER * 8);
  float* denom    = carve((size_t)N_PAPER * 8);
  float* colsum_w = carve(HID);
  float* colsum_c = carve(HID);
  float* attn     = carve(2);
  (void)ws_size; (void)n_in; (void)out_size;

  const dim3 B256(256);

  // 1) node-type projections (WMMA f32)
  proj_gemm<<<N_PAPER / 16, B256, 0, stream>>>(x_paper, W_paper, b_paper, h_paper);
  proj_gemm<<<N_AUTHOR / 16, B256, 0, stream>>>(x_author, W_author, b_author, h_author);

  // 2) per-node attention logits
  node_att<<<(N_AUTHOR * 8 + 255) / 256, B256, 0, stream>>>(h_author, att_s_w, a_s_wr, N_AUTHOR * 8);
  node_att<<<(N_PAPER * 8 + 255) / 256, B256, 0, stream>>>(h_paper, att_d_w, a_d_wr, N_PAPER * 8);
  node_att<<<(N_PAPER * 8 + 255) / 256, B256, 0, stream>>>(h_paper, att_s_c, a_s_ci, N_PAPER * 8);
  node_att<<<(N_PAPER * 8 + 255) / 256, B256, 0, stream>>>(h_paper, att_d_c, a_d_ci, N_PAPER * 8);

  const size_t segBytes = (size_t)N_PAPER * 8 * sizeof(float);
  const size_t outBytes = (size_t)N_PAPER * HID * sizeof(float);

  // 3) writes relation: author -> paper
  hipMemsetAsync(maxb, 0xFF, segBytes, stream);   // -NaN: dominated by any real value
  hipMemsetAsync(denom, 0, segBytes, stream);
  hipMemsetAsync(out_w, 0, outBytes, stream);
  edge_max<<<(E_W + 255) / 256, B256, 0, stream>>>(a_s_wr, a_d_wr, writes_src, writes_dst, maxb, E_W);
  edge_sum<<<(E_W + 255) / 256, B256, 0, stream>>>(a_s_wr, a_d_wr, writes_src, writes_dst, maxb, denom, E_W);
  edge_scatter<<<(E_W + 7) / 8, B256, 0, stream>>>(a_s_wr, a_d_wr, writes_src, writes_dst,
                                                   maxb, denom, h_author, out_w, E_W);
  relu_inplace<<<(N_PAPER * HID + 255) / 256, B256, 0, stream>>>(out_w, N_PAPER * HID);

  // 4) cites relation: paper -> paper (reuse max/denom buffers)
  hipMemsetAsync(maxb, 0xFF, segBytes, stream);
  hipMemsetAsync(denom, 0, segBytes, stream);
  hipMemsetAsync(out_c, 0, outBytes, stream);
  edge_max<<<(E_C + 255) / 256, B256, 0, stream>>>(a_s_ci, a_d_ci, cites_src, cites_dst, maxb, E_C);
  edge_sum<<<(E_C + 255) / 256, B256, 0, stream>>>(a_s_ci, a_d_ci, cites_src, cites_dst, maxb, denom, E_C);
  edge_scatter<<<(E_C + 7) / 8, B256, 0, stream>>>(a_s_ci, a_d_ci, cites_src, cites_dst,
                                                   maxb, denom, h_paper, out_c, E_C);
  relu_inplace<<<(N_PAPER * HID + 255) / 256, B256, 0, stream>>>(out_c, N_PAPER * HID);

  // 5) semantic attention scores (fused GEMM + tanh + column-sum)
  hipMemsetAsync(colsum_w, 0, HID * sizeof(float), stream);
  hipMemsetAsync(colsum_c, 0, HID * sizeof(float), stream);
  sem_gemm<<<N_PAPER / 16, B256, 0, stream>>>(out_w, k_W, k_b, colsum_w);
  sem_gemm<<<N_PAPER / 16, B256, 0, stream>>>(out_c, k_W, k_b, colsum_c);
  sem_score<<<1, 32, 0, stream>>>(q, colsum_w, colsum_c, attn);

  // 6) final: (a0*out_w + a1*out_c) @ lin_W + lin_b  (combine fused into A-load)
  const int nTiles = (N_PAPER / 16) * (EMB / 16);
  final_gemm<<<(nTiles + 7) / 8, B256, 0, stream>>>(out_w, out_c, attn, lin_W, lin_b,
                                                    (float*)d_out);
}